// QuantizedBn2d_69518340653524
// MI455X (gfx1250) — compile-verified
//
#include <hip/hip_runtime.h>

typedef float v4f __attribute__((ext_vector_type(4)));

// Exact gemmlowp requantization for one element.
// subsum fits in ~18 bits, so the <<sneg pre-shift (sneg<=2) is exact in i32,
// the i64 work is one sext 32x32->64 multiply + one 64-bit add, and the
// (ab+nudge)>>31 result fits in i32 so the rounding shift runs fully 32-bit.
__device__ __forceinline__ float requant_one(float xf, int A, int K,
                                             int sneg, int spos, int maski,
                                             int thrbase, int m0, int Z3)
{
    int xi     = (int)xf;                              // activations are exact small ints
    int subsum = xi * A + K;                           // int32 (reference computes in int32)
    int ss     = (int)((unsigned)subsum << sneg);      // exact pre-shift for negative shift
    long long ab    = (long long)ss * (long long)m0;   // v_mad_i64_i32 path
    long long nudge = (ab >= 0) ? (1LL << 30) : (1LL - (1LL << 30));
    int mult  = (int)((ab + nudge) >> 31);             // |mult| <= 2^20 -> i32 safe
    int rem   = mult & maski;
    int thr   = thrbase + ((mult < 0) ? 1 : 0);
    int total = (mult >> spos) + ((rem > thr) ? 1 : 0) + Z3;
    total = total < -128 ? -128 : (total > 127 ? 127 : total);
    return (float)total;
}

// 2D grid: blockIdx.x = channel, blockIdx.y = batch. One workgroup per
// (b, c) plane: all quantization scalars are block-uniform -> SMEM/SALU
// setup once (no integer division anywhere), pure streaming loop after.
extern "C" __global__ void __launch_bounds__(256)
qbn2d_pcq_kernel(const float* __restrict__ x,
                 const int*   __restrict__ weight,   // [NCL, C]
                 const int*   __restrict__ bias,     // [NCL, C]
                 const int*   __restrict__ z1,
                 const int*   __restrict__ z2,
                 const int*   __restrict__ z3,
                 const int*   __restrict__ M0,
                 const int*   __restrict__ shiftv,
                 const int*   __restrict__ bcl,      // [B]
                 float*       __restrict__ out,
                 int HW)
{
    const int c  = blockIdx.x;
    const int b  = blockIdx.y;
    const int C  = gridDim.x;
    const int cl = bcl[b];

    const int w   = weight[cl * C + c];
    const int bs  = bias[cl * C + c];
    const int Z1  = z1[cl];
    const int Z2  = z2[cl];
    const int Z3  = z3[cl];
    const int m0  = M0[cl];
    const int sh  = shiftv[cl];

    const int A       = w - Z2;                 // multiplier on xi
    const int K       = Z1 * Z2 - w * Z1 + bs;  // block-uniform additive constant
    const int sneg    = sh < 0 ? -sh : 0;       // shift<0 -> exact pre-left-shift
    const int spos    = sh > 0 ?  sh : 0;       // shift>0 -> rounding right shift
    const int maski   = (1 << spos) - 1;        // spos <= 11, fits i32
    const int thrbase = maski >> 1;

    const size_t base = ((size_t)b * (size_t)C + (size_t)c) * (size_t)HW;
    const v4f* __restrict__ xin = (const v4f*)(x + base);
    v4f*       __restrict__ po  = (v4f*)(out + base);
    const int n4 = HW >> 2;

    // Streaming loop: b128 NT loads/stores (each byte touched exactly once;
    // 205MB total > 192MB L2 so NT avoids thrashing). Prefetch is issued
    // unconditionally: speculative prefetches past the plane are silently
    // dropped by hardware on translation failure, so no branch needed.
    for (int i = threadIdx.x; i < n4; i += 256) {
        __builtin_prefetch(&xin[i + 256], 0, 0);       // global_prefetch_b8
        v4f v = __builtin_nontemporal_load(&xin[i]);   // global_load_b128 (NT)
        v4f o;
        o.x = requant_one(v.x, A, K, sneg, spos, maski, thrbase, m0, Z3);
        o.y = requant_one(v.y, A, K, sneg, spos, maski, thrbase, m0, Z3);
        o.z = requant_one(v.z, A, K, sneg, spos, maski, thrbase, m0, Z3);
        o.w = requant_one(v.w, A, K, sneg, spos, maski, thrbase, m0, Z3);
        __builtin_nontemporal_store(o, &po[i]);        // global_store_b128 (NT)
    }

    // Tail (HW not divisible by 4) — not taken for 56x56 but kept for safety.
    const int rem = HW & 3;
    if ((int)threadIdx.x < rem) {
        const int idx = (n4 << 2) + (int)threadIdx.x;
        out[base + idx] = requant_one(x[base + idx], A, K, sneg, spos, maski,
                                      thrbase, m0, Z3);
    }
}

extern "C" void kernel_launch(void* const* d_in, const int* in_sizes, int n_in,
                              void* d_out, int out_size, void* d_ws, size_t ws_size,
                              hipStream_t stream)
{
    const float* x      = (const float*)d_in[0];
    const int*   weight = (const int*)d_in[1];
    const int*   bias   = (const int*)d_in[2];
    const int*   z1     = (const int*)d_in[3];
    const int*   z2     = (const int*)d_in[4];
    const int*   z3     = (const int*)d_in[5];
    const int*   M0     = (const int*)d_in[6];
    const int*   shift  = (const int*)d_in[7];
    const int*   bcl    = (const int*)d_in[8];
    float*       out    = (float*)d_out;

    const int B   = in_sizes[8];              // len(batch_cluster)
    const int NCL = in_sizes[3];              // len(z1)
    const int C   = in_sizes[1] / NCL;        // weight is [NCL, C]
    const int HW  = in_sizes[0] / (B * C);    // x is [B, C, H, W]

    dim3 grid(C, B);                          // no division in-kernel
    dim3 block(256);                          // 8 wave32 waves
    qbn2d_pcq_kernel<<<grid, block, 0, stream>>>(
        x, weight, bias, z1, z2, z3, M0, shift, bcl, out, HW);
}